// LEConvNet_1layer_88553635709230
// MI455X (gfx1250) — compile-verified
//
#include <hip/hip_runtime.h>
#include <hip/hip_bf16.h>
#include <math.h>

typedef __attribute__((ext_vector_type(2))) float v2f;
typedef __attribute__((ext_vector_type(8))) float v8f;

#define F_IN   128
#define F_HID  32
#define NCLS   10
#define WSTRIDE 40   // padded LDS row stride: 2*40 mod 64 = 16 -> lane halves hit disjoint banks

// ---------------- Kernel 0: zero the atomic-accumulation scratch ----------------
__global__ __launch_bounds__(256) void zero_ws(float* __restrict__ p, long n) {
    long i = (long)blockIdx.x * blockDim.x + threadIdx.x;
    if (i < n) p[i] = 0.0f;
}

// ---------------- Kernel 1: fused 3-way GEMM via V_WMMA_F32_16X16X4_F32 ----------
// a = x@W1 + b1 ; b = x@W2 ; c = x@W3 + b3       (x: [N,128], W*: [128,32])
// One wave per 16-row tile; 6 column-tiles (W1 lo/hi, W2 lo/hi, W3 lo/hi).
__global__ __launch_bounds__(256) void gemm3_wmma(
        const float* __restrict__ x,
        const float* __restrict__ W1, const float* __restrict__ b1,
        const float* __restrict__ W2,
        const float* __restrict__ W3, const float* __restrict__ b3,
        float* __restrict__ abuf, float* __restrict__ bbuf, float* __restrict__ cbuf,
        int N, int ntiles) {
    __shared__ float Wlds[3 * F_IN * WSTRIDE];   // 60 KB

    // Stage W1|W2|W3 into padded LDS (every wave in the block reuses them).
    const float* Ws[3] = {W1, W2, W3};
    #pragma unroll
    for (int w = 0; w < 3; ++w) {
        const float* s = Ws[w];
        for (int t = threadIdx.x; t < F_IN * F_HID; t += blockDim.x) {
            int row = t >> 5, col = t & 31;
            Wlds[w * F_IN * WSTRIDE + row * WSTRIDE + col] = s[t];
        }
    }
    __syncthreads();

    const int lane = threadIdx.x & 31;
    const int wid  = threadIdx.x >> 5;
    const int tile = blockIdx.x * (blockDim.x >> 5) + wid;
    if (tile >= ntiles) return;                  // wave-uniform: EXEC stays all-ones below
    const int m0 = tile * 16;

    // A-fragment addressing (ISA 16x4 f32 layout): lanes 0-15 -> K={0,1}, lanes 16-31 -> K={2,3}
    int arow = m0 + (lane & 15);
    if (arow >= N) arow = N - 1;                 // clamp (N%16==0 in practice)
    const int   ksel = (lane >> 4) * 2;
    const float* xrow = x + (size_t)arow * F_IN;

    v8f acc[6] = {};                             // 6 column tiles x 16x16 f32

    #pragma unroll 4
    for (int ks = 0; ks < F_IN / 4; ++ks) {      // 32 K-steps of 4
        const int kb = 4 * ks + ksel;
        v2f afrag;
        afrag.x = xrow[kb];
        afrag.y = xrow[kb + 1];
        #pragma unroll
        for (int ct = 0; ct < 6; ++ct) {
            // B-fragment (4x16): lanes 0-15 rows {kb,kb+1}, lanes 16-31 rows {kb+2,kb+3} via ksel
            const float* wl = &Wlds[(ct >> 1) * F_IN * WSTRIDE + kb * WSTRIDE
                                    + (ct & 1) * 16 + (lane & 15)];
            v2f bfrag;
            bfrag.x = wl[0];
            bfrag.y = wl[WSTRIDE];
            acc[ct] = __builtin_amdgcn_wmma_f32_16x16x4_f32(
                false, afrag, false, bfrag, (short)0, acc[ct], false, false);
        }
    }

    // C/D layout: VGPR r -> row r (lanes 0-15) / row r+8 (lanes 16-31), col = lane&15
    const int col   = lane & 15;
    const int roff  = (lane >> 4) * 8;
    #pragma unroll
    for (int r = 0; r < 8; ++r) {
        const int row = m0 + roff + r;
        if (row < N) {
            float* ap = abuf + (size_t)row * F_HID;
            float* bp = bbuf + (size_t)row * F_HID;
            float* cp = cbuf + (size_t)row * F_HID;
            ap[col]      = acc[0][r] + b1[col];
            ap[col + 16] = acc[1][r] + b1[col + 16];
            bp[col]      = acc[2][r];
            bp[col + 16] = acc[3][r];
            cp[col]      = acc[4][r] + b3[col];
            cp[col + 16] = acc[5][r] + b3[col + 16];
        }
    }
}

// ---------------- Kernel 2: edge scatter (wave per edge, lane = feature) ---------
// degw[d] += ew ; sb[d][lane] += ew * b[s][lane]   -> one coalesced 128B atomic line/edge
__global__ __launch_bounds__(256) void edge_scatter(
        const int* __restrict__ srcIdx, const int* __restrict__ dstIdx,
        const float* __restrict__ ea, const float* __restrict__ bbuf,
        float* __restrict__ degw, float* __restrict__ sb, int E) {
    long tid  = (long)blockIdx.x * blockDim.x + threadIdx.x;
    int  e    = (int)(tid >> 5);
    int  lane = (int)(tid & 31);
    if (e >= E) return;
    int   s = srcIdx[e];
    int   d = dstIdx[e];
    float w = ea[e];
    float bv = bbuf[(size_t)s * F_HID + lane];
    atomicAdd(&sb[(size_t)d * F_HID + lane], w * bv);
    if (lane == 0) atomicAdd(&degw[d], w);
}

// ---------------- Kernel 3: per-node epilogue: combine + ELU + FC + log_softmax --
__global__ __launch_bounds__(256) void node_final(
        const float* __restrict__ abuf, const float* __restrict__ cbuf,
        const float* __restrict__ sb,   const float* __restrict__ degw,
        const float* __restrict__ fcw,  const float* __restrict__ fcb,
        float* __restrict__ out, int N) {
    int i = blockIdx.x * blockDim.x + threadIdx.x;
    if (i >= N) return;
    float dw = degw[i];

    float logit[NCLS];
    #pragma unroll
    for (int c = 0; c < NCLS; ++c) logit[c] = fcb[c];

    const float4* a4 = (const float4*)(abuf + (size_t)i * F_HID);
    const float4* c4 = (const float4*)(cbuf + (size_t)i * F_HID);
    const float4* s4 = (const float4*)(sb   + (size_t)i * F_HID);
    #pragma unroll
    for (int j = 0; j < F_HID / 4; ++j) {
        float4 av = a4[j], cv = c4[j], sv = s4[j];
        float h[4] = { av.x * dw - sv.x + cv.x,  av.y * dw - sv.y + cv.y,
                       av.z * dw - sv.z + cv.z,  av.w * dw - sv.w + cv.w };
        #pragma unroll
        for (int t = 0; t < 4; ++t) h[t] = (h[t] > 0.0f) ? h[t] : (expf(h[t]) - 1.0f);
        #pragma unroll
        for (int c = 0; c < NCLS; ++c) {
            const float* wr = fcw + c * F_HID + j * 4;   // uniform address -> scalar loads
            logit[c] += h[0] * wr[0] + h[1] * wr[1] + h[2] * wr[2] + h[3] * wr[3];
        }
    }
    float m = logit[0];
    #pragma unroll
    for (int c = 1; c < NCLS; ++c) m = fmaxf(m, logit[c]);
    float ssum = 0.0f;
    #pragma unroll
    for (int c = 0; c < NCLS; ++c) ssum += expf(logit[c] - m);
    float lse = m + logf(ssum);
    float* o = out + (size_t)i * NCLS;
    #pragma unroll
    for (int c = 0; c < NCLS; ++c) o[c] = logit[c] - lse;
}

// ---------------------------------- launcher -------------------------------------
extern "C" void kernel_launch(void* const* d_in, const int* in_sizes, int n_in,
                              void* d_out, int out_size, void* d_ws, size_t ws_size,
                              hipStream_t stream) {
    const float* x   = (const float*)d_in[0];
    const int*   ei  = (const int*)  d_in[1];   // [2,E] : row0 = src, row1 = dst
    const float* ea  = (const float*)d_in[2];
    const float* W1  = (const float*)d_in[3];
    const float* b1  = (const float*)d_in[4];
    const float* W2  = (const float*)d_in[5];
    const float* W3  = (const float*)d_in[6];
    const float* b3  = (const float*)d_in[7];
    const float* fcw = (const float*)d_in[8];
    const float* fcb = (const float*)d_in[9];
    float* out = (float*)d_out;

    const int N = in_sizes[0] / F_IN;
    const int E = in_sizes[2];
    const int* srcIdx = ei;
    const int* dstIdx = ei + E;

    // workspace layout (floats): a[N*32] | b[N*32] | c[N*32] | degw[N] | sb[N*32]
    float* abuf = (float*)d_ws;
    float* bbuf = abuf + (size_t)N * F_HID;
    float* cbuf = bbuf + (size_t)N * F_HID;
    float* degw = cbuf + (size_t)N * F_HID;
    float* sbuf = degw + N;

    long nz = (long)N * (F_HID + 1);            // degw + sb are contiguous
    zero_ws<<<(int)((nz + 255) / 256), 256, 0, stream>>>(degw, nz);

    const int ntiles = (N + 15) / 16;
    gemm3_wmma<<<(ntiles + 7) / 8, 256, 0, stream>>>(x, W1, b1, W2, W3, b3,
                                                     abuf, bbuf, cbuf, N, ntiles);

    long et = (long)E * 32;
    edge_scatter<<<(int)((et + 255) / 256), 256, 0, stream>>>(srcIdx, dstIdx, ea,
                                                              bbuf, degw, sbuf, E);

    node_final<<<(N + 255) / 256, 256, 0, stream>>>(abuf, cbuf, sbuf, degw,
                                                    fcw, fcb, out, N);
}